// DownsampleLayer_80418967650322
// MI455X (gfx1250) — compile-verified
//
#include <hip/hip_runtime.h>

// ---------- types ----------
typedef __attribute__((ext_vector_type(2))) float v2f;
typedef __attribute__((ext_vector_type(8))) float v8f;

static __device__ __forceinline__ v8f wmma_f32_4(v2f a, v2f b, v8f c) {
    // D = A(16x4) * B(4x16) + C(16x16), full fp32
    return __builtin_amdgcn_wmma_f32_16x16x4_f32(
        /*neg_a=*/false, a, /*neg_b=*/false, b,
        /*c_mod=*/(short)0, c, /*reuse_a=*/false, /*reuse_b=*/false);
}

// ---------- weight pre-transpose into WMMA lane layout ----------
// Wt flat index o = (((k*NK + q)*NT + t)*32 + lane)*2 + e
//   maps to W[(k*CIN + (q*4 + (lane>>4)*2 + e))*COUT + (t*16 + (lane&15))]
__global__ void transpose_w_kernel(const float* __restrict__ W, float* __restrict__ Wt,
                                   int CIN, int COUT, long total) {
    long o = (long)blockIdx.x * blockDim.x + threadIdx.x;
    if (o >= total) return;
    const int NK = CIN >> 2, NT = COUT >> 4;
    const int e    = (int)(o & 1);
    const int lane = (int)((o >> 1) & 31);
    long rem = o >> 6;
    const int t = (int)(rem % NT); rem /= NT;
    const int q = (int)(rem % NK);
    const long k = rem / NK;
    const int c = q * 4 + ((lane >> 4) << 1) + e;
    const int d = t * 16 + (lane & 15);
    Wt[o] = W[(k * CIN + c) * COUT + d];
}

// ---------- vectorized elementwise helpers ----------
__global__ void fill_bias4_kernel(float4* __restrict__ out, const float4* __restrict__ b,
                                  int c4mask, long total4) {
    long i = (long)blockIdx.x * blockDim.x + threadIdx.x;
    if (i < total4) out[i] = b[i & c4mask];
}

__global__ void relu4_kernel(float4* __restrict__ x, long total4) {
    long i = (long)blockIdx.x * blockDim.x + threadIdx.x;
    if (i < total4) {
        float4 v = x[i];
        v.x = fmaxf(v.x, 0.f); v.y = fmaxf(v.y, 0.f);
        v.z = fmaxf(v.z, 0.f); v.w = fmaxf(v.w, 0.f);
        x[i] = v;
    }
}

// out[r, 0:32] = p0[r, :] + xres[r, 0:32]   (float4 lanes: 8 per row)
__global__ void add_p0_4_kernel(const float4* __restrict__ p0, const float4* __restrict__ xres,
                                float4* __restrict__ out, int n) {
    long i = (long)blockIdx.x * blockDim.x + threadIdx.x;
    if (i < (long)n * 8) {
        long r = i >> 3;
        int c4 = (int)(i & 7);
        float4 a = p0[i], b = xres[r * 16 + c4];
        out[r * 16 + c4] = make_float4(a.x + b.x, a.y + b.y, a.z + b.z, a.w + b.w);
    }
}

// ---------- sparse conv: branch-free gather - WMMA - atomic scatter ----------
// x:[n_in,CIN], Wt: transposed weights, idx:[K,P]; grid=(ceil(P/16),K), block=32
template <int CIN, int COUT, bool RELU_IN>
__global__ __launch_bounds__(32)
void sconv_kernel(const float* __restrict__ x, const float* __restrict__ Wt,
                  const int* __restrict__ in_idx, const int* __restrict__ out_idx,
                  float* __restrict__ out, int n_in, int n_out, int P) {
    constexpr int NK = CIN / 4, NT = COUT / 16;
    const int k    = blockIdx.y;
    const int p0   = blockIdx.x * 16;
    const int lane = threadIdx.x;
    const int m    = lane & 15;
    const int koff = (lane >> 4) * 2;

    const int* ii = in_idx  + (long)k * P + p0;
    const int* oi = out_idx + (long)k * P + p0;

    // pairs are compacted: if first row of tile is sentinel, whole tile is dead
    if (oi[0] >= n_out) return;

    int irow, orow;
    if (p0 + m < P) { irow = ii[m]; orow = oi[m]; }
    else            { irow = n_in;  orow = n_out; }

    // branch-free gather: clamp sentinel rows to row 0 and zero them after load
    const bool valid = (irow < n_in);
    const float sel  = valid ? 1.0f : 0.0f;
    const float* __restrict__ xrow = x + (long)(valid ? irow : 0) * CIN + koff;

    v2f A[NK];
#pragma unroll
    for (int q = 0; q < NK; ++q) A[q] = *(const v2f*)(xrow + q * 4);
#pragma unroll
    for (int q = 0; q < NK; ++q) {
        float ax = A[q].x * sel, ay = A[q].y * sel;
        if (RELU_IN) { ax = fmaxf(ax, 0.f); ay = fmaxf(ay, 0.f); }
        A[q].x = ax; A[q].y = ay;
    }

    const v2f* __restrict__ wt = (const v2f*)Wt + (long)k * NK * NT * 32 + lane;

    v8f acc[NT];
#pragma unroll
    for (int t = 0; t < NT; ++t) acc[t] = (v8f){0, 0, 0, 0, 0, 0, 0, 0};
#pragma unroll
    for (int q = 0; q < NK; ++q)
#pragma unroll
        for (int t = 0; t < NT; ++t)
            acc[t] = wmma_f32_4(A[q], wt[(q * NT + t) * 32], acc[t]);

    // scatter: D layout -> VGPR j holds row (j + (lane>>4)*8), col (lane&15)
    const int rbase = (lane >> 4) * 8;
    const int colb  = lane & 15;
#pragma unroll
    for (int j = 0; j < 8; ++j) {
        const int ro = __shfl(orow, rbase + j, 32);
        if (ro < n_out) {
            float* op = out + (long)ro * COUT + colb;
#pragma unroll
            for (int t = 0; t < NT; ++t)
                unsafeAtomicAdd(op + t * 16, acc[t][j]);
        }
    }
}

// ---------- dense GEMM: p1a = relu(x[n,64] @ W10[64,16] + b10) ----------
__global__ __launch_bounds__(32)
void gemm16_relu_kernel(const float* __restrict__ x, const float* __restrict__ Wt,
                        const float* __restrict__ bias, float* __restrict__ out, int n) {
    constexpr int NK = 16;  // CIN=64
    const int lane = threadIdx.x;
    const int m    = lane & 15;
    const int koff = (lane >> 4) * 2;
    const int r    = blockIdx.x * 16 + m;
    const bool valid = (r < n);
    const float sel  = valid ? 1.0f : 0.0f;
    const float* __restrict__ xrow = x + (long)(valid ? r : 0) * 64 + koff;
    const v2f* __restrict__ wt = (const v2f*)Wt + lane;

    v2f A[NK];
#pragma unroll
    for (int q = 0; q < NK; ++q) A[q] = *(const v2f*)(xrow + q * 4);
#pragma unroll
    for (int q = 0; q < NK; ++q) { A[q].x *= sel; A[q].y *= sel; }

    v8f acc = (v8f){0, 0, 0, 0, 0, 0, 0, 0};
#pragma unroll
    for (int q = 0; q < NK; ++q) acc = wmma_f32_4(A[q], wt[q * 32], acc);

    const int rbase = (lane >> 4) * 8;
    const int colb  = lane & 15;
#pragma unroll
    for (int j = 0; j < 8; ++j) {
        const int ro = blockIdx.x * 16 + rbase + j;
        if (ro < n)
            out[(long)ro * 16 + colb] = fmaxf(acc[j] + bias[colb], 0.0f);
    }
}

// ---------- final: out[r,32:64] = relu(p1b)[n,16] @ W12[16,32] + b12 + xres[r,32:64] ----------
__global__ __launch_bounds__(32)
void gemm_final_kernel(const float* __restrict__ p1b, const float* __restrict__ Wt,
                       const float* __restrict__ bias, const float* __restrict__ xres,
                       float* __restrict__ out, int n) {
    constexpr int NK = 4, NT = 2;  // CIN=16, COUT=32
    const int lane = threadIdx.x;
    const int m    = lane & 15;
    const int koff = (lane >> 4) * 2;
    const int r    = blockIdx.x * 16 + m;
    const bool valid = (r < n);
    const float sel  = valid ? 1.0f : 0.0f;
    const float* __restrict__ xrow = p1b + (long)(valid ? r : 0) * 16 + koff;
    const v2f* __restrict__ wt = (const v2f*)Wt + lane;

    v2f A[NK];
#pragma unroll
    for (int q = 0; q < NK; ++q) A[q] = *(const v2f*)(xrow + q * 4);
#pragma unroll
    for (int q = 0; q < NK; ++q) {  // fused relu(p1b) + sentinel zero
        A[q].x = fmaxf(A[q].x * sel, 0.f);
        A[q].y = fmaxf(A[q].y * sel, 0.f);
    }

    v8f acc[NT];
#pragma unroll
    for (int t = 0; t < NT; ++t) acc[t] = (v8f){0, 0, 0, 0, 0, 0, 0, 0};
#pragma unroll
    for (int q = 0; q < NK; ++q)
#pragma unroll
        for (int t = 0; t < NT; ++t)
            acc[t] = wmma_f32_4(A[q], wt[(q * NT + t) * 32], acc[t]);

    const int rbase = (lane >> 4) * 8;
    const int colb  = lane & 15;
#pragma unroll
    for (int j = 0; j < 8; ++j) {
        const int ro = blockIdx.x * 16 + rbase + j;
        if (ro < n) {
            const long base = (long)ro * 64 + 32;
#pragma unroll
            for (int t = 0; t < NT; ++t) {
                const int c = colb + t * 16;
                out[base + c] = acc[t][j] + bias[c] + xres[base + c];
            }
        }
    }
}

// ---------- host orchestration ----------
extern "C" void kernel_launch(void* const* d_in, const int* in_sizes, int n_in_args,
                              void* d_out, int out_size, void* d_ws, size_t ws_size,
                              hipStream_t stream) {
    (void)n_in_args; (void)ws_size;
    const float* feats  = (const float*)d_in[0];
    const float* W_conv = (const float*)d_in[1];
    const float* b_conv = (const float*)d_in[2];
    const float* W_down = (const float*)d_in[3];
    const float* b_down = (const float*)d_in[4];
    const float* W00 = (const float*)d_in[5];
    const float* b00 = (const float*)d_in[6];
    const float* W01 = (const float*)d_in[7];
    const float* b01 = (const float*)d_in[8];
    const float* W10 = (const float*)d_in[9];
    const float* b10 = (const float*)d_in[10];
    const float* W11 = (const float*)d_in[11];
    const float* b11 = (const float*)d_in[12];
    const float* W12 = (const float*)d_in[13];
    const float* b12 = (const float*)d_in[14];
    const int* in_idx1  = (const int*)d_in[15];
    const int* out_idx1 = (const int*)d_in[16];
    const int* in_idxd  = (const int*)d_in[17];
    const int* out_idxd = (const int*)d_in[18];
    const int* in_idx2  = (const int*)d_in[19];
    const int* out_idx2 = (const int*)d_in[20];

    const int N      = in_sizes[0] / 32;     // input points
    const int P1     = in_sizes[15] / 27;    // kmap pad, layer1
    const int Pd     = in_sizes[17] / 8;     // kmap pad, downsample
    const int P2     = in_sizes[19] / 27;    // kmap pad, stride-2 convs
    const int n_down = out_size / 64;        // output rows (host-known)

    // workspace layout (fp32)
    float* h   = (float*)d_ws;                 // [N, 64]
    float* xb  = h   + (long)N * 64;           // [n_down, 64]  (y -> relu -> x)
    float* p0a = xb  + (long)n_down * 64;      // [n_down, 16]
    float* p0b = p0a + (long)n_down * 16;      // [n_down, 32]
    float* p1a = p0b + (long)n_down * 32;      // [n_down, 16]
    float* p1b = p1a + (long)n_down * 16;      // [n_down, 16]
    float* wt_conv = p1b + (long)n_down * 16;  // transposed weights
    float* wt_down = wt_conv + 27L * 32 * 64;
    float* wt00 = wt_down + 8L * 64 * 64;
    float* wt01 = wt00 + 27L * 64 * 16;
    float* wt10 = wt01 + 27L * 16 * 32;
    float* wt11 = wt10 + 64L * 16;
    float* wt12 = wt11 + 27L * 16 * 16;
    float* outf = (float*)d_out;               // [n_down, 64]

    const dim3 B32(32);
    const dim3 B256(256);
    auto ceil16  = [](int v) { return (unsigned)((v + 15) / 16); };
    auto ceil256 = [](long v) { return (unsigned)((v + 255) / 256); };

    // ---- weight transposes (tiny)
    auto xpose = [&](const float* W, float* Wt, int CIN, int COUT, long K) {
        long tot = K * CIN * COUT;
        transpose_w_kernel<<<ceil256(tot), B256, 0, stream>>>(W, Wt, CIN, COUT, tot);
    };
    xpose(W_conv, wt_conv, 32, 64, 27);
    xpose(W_down, wt_down, 64, 64, 8);
    xpose(W00, wt00, 64, 16, 27);
    xpose(W01, wt01, 16, 32, 27);
    xpose(W10, wt10, 64, 16, 1);
    xpose(W11, wt11, 16, 16, 27);
    xpose(W12, wt12, 16, 32, 1);

    // ---- layer 1: h = sconv(feats, W_conv) + b_conv  (relu folded into consumer)
    long hN = (long)N * 64;
    fill_bias4_kernel<<<ceil256(hN / 4), B256, 0, stream>>>(
        (float4*)h, (const float4*)b_conv, 15, hN / 4);
    sconv_kernel<32, 64, false><<<dim3(ceil16(P1), 27), B32, 0, stream>>>(
        feats, wt_conv, in_idx1, out_idx1, h, N, N, P1);

    // ---- downsample: x = relu(sconv(relu(h), W_down) + b_down)
    long xN = (long)n_down * 64;
    fill_bias4_kernel<<<ceil256(xN / 4), B256, 0, stream>>>(
        (float4*)xb, (const float4*)b_down, 15, xN / 4);
    sconv_kernel<64, 64, true><<<dim3(ceil16(Pd), 8), B32, 0, stream>>>(
        h, wt_down, in_idxd, out_idxd, xb, N, n_down, Pd);
    relu4_kernel<<<ceil256(xN / 4), B256, 0, stream>>>((float4*)xb, xN / 4);

    // ---- path 0: p0a = sconv(x, W00)+b00 ; p0b = sconv(relu(p0a), W01)+b01
    long aN = (long)n_down * 16;
    fill_bias4_kernel<<<ceil256(aN / 4), B256, 0, stream>>>(
        (float4*)p0a, (const float4*)b00, 3, aN / 4);
    sconv_kernel<64, 16, false><<<dim3(ceil16(P2), 27), B32, 0, stream>>>(
        xb, wt00, in_idx2, out_idx2, p0a, n_down, n_down, P2);

    long bN = (long)n_down * 32;
    fill_bias4_kernel<<<ceil256(bN / 4), B256, 0, stream>>>(
        (float4*)p0b, (const float4*)b01, 7, bN / 4);
    sconv_kernel<16, 32, true><<<dim3(ceil16(P2), 27), B32, 0, stream>>>(
        p0a, wt01, in_idx2, out_idx2, p0b, n_down, n_down, P2);

    // ---- path 1: p1a = relu(x @ W10 + b10); p1b = sconv(p1a, W11)+b11
    gemm16_relu_kernel<<<ceil16(n_down), B32, 0, stream>>>(xb, wt10, b10, p1a, n_down);

    fill_bias4_kernel<<<ceil256(aN / 4), B256, 0, stream>>>(
        (float4*)p1b, (const float4*)b11, 3, aN / 4);
    sconv_kernel<16, 16, false><<<dim3(ceil16(P2), 27), B32, 0, stream>>>(
        p1a, wt11, in_idx2, out_idx2, p1b, n_down, n_down, P2);

    // ---- epilogue: out[:,32:64] = relu(p1b) @ W12 + b12 + x[:,32:64]
    gemm_final_kernel<<<ceil16(n_down), B32, 0, stream>>>(p1b, wt12, b12, xb, outf, n_down);
    // ---- out[:,0:32] = p0b + x[:,0:32]
    add_p0_4_kernel<<<ceil256(bN / 4), B256, 0, stream>>>(
        (const float4*)p0b, (const float4*)xb, (float4*)outf, n_down);
}